// IAFM_58944131170777
// MI455X (gfx1250) — compile-verified
//
#include <hip/hip_runtime.h>

typedef __attribute__((ext_vector_type(2))) float v2f;
typedef __attribute__((ext_vector_type(8))) float v8f;

namespace {
constexpr int kN   = 20;   // N0 == N1 == 20
constexpr int kD   = 64;   // embedding dim
constexpr int kDP  = 66;   // padded LDS row stride (bank-conflict-free WMMA loads)
constexpr int kBatchesPerPair = 16;       // one WMMA tile of batches
constexpr int kPairsPerBlock  = 4;        // 8 waves / block
constexpr int kThreads        = kPairsPerBlock * 64;                 // 256
constexpr int kBatchesPerBlock = kPairsPerBlock * kBatchesPerPair;   // 64
}

__global__ __launch_bounds__(kThreads)
void iafm_kernel(const int*   __restrict__ ii0,
                 const int*   __restrict__ ii1,
                 const float* __restrict__ vecs,
                 const float* __restrict__ nb,
                 float*       __restrict__ out,
                 int B)
{
    // [pair][side][batch][d(padded)] : 4*2*16*66*4 = 33 KB
    __shared__ float smem[kPairsPerBlock][2][kBatchesPerPair][kDP];

    const int lane = threadIdx.x & 31;
    const int wave = threadIdx.x >> 5;
    const int pair = wave >> 1;
    const int side = wave & 1;          // wave0: ii0 -> S0, wave1: ii1 -> S1
    const int bb   = blockIdx.x * kBatchesPerBlock + pair * kBatchesPerPair;

    const int* idxbase = side ? ii1 : ii0;
    float (*S)[kDP] = smem[pair][side];

    // ---- Phase 1: gather + row-sum. Lane l owns d = 2l, 2l+1 (float2). ----
    // One global_load_b64 per row = one fully coalesced 256B embedding row.
    for (int i = 0; i < kBatchesPerPair; ++i) {
        int b = bb + i;
        if (b >= B) b = B - 1;                       // clamp (dup work, store guarded)
        int myidx = (lane < kN) ? idxbase[b * kN + lane] : 0;
        v2f acc = {0.f, 0.f};
#pragma unroll
        for (int r = 0; r < kN; ++r) {
            int idx = __builtin_amdgcn_readlane(myidx, r);   // uniform -> SGPR base
            v2f v = *(const v2f*)(vecs + (size_t)idx * kD + 2 * lane);
            acc += v;                                        // v_pk_add_f32
        }
        *(v2f*)&S[i][2 * lane] = acc;                        // ds_store_b64
    }

    __syncthreads();

    // ---- Phase 2 (even waves): diag(S0 * S1^T) via 16x V_WMMA_F32_16X16X4_F32 ----
    if (side == 0) {
        const float (*S0)[kDP] = smem[pair][0];
        const float (*S1)[kDP] = smem[pair][1];
        const int m    = lane & 15;             // A: row M / B: col N
        const int koff = (lane >> 4) * 2;       // half-wave K split per ISA layout
        v8f C = {};
#pragma unroll
        for (int c = 0; c < kD / 4; ++c) {
            int k = 4 * c + koff;               // VGPR0->K,K+1 maps via v2f lanes
            v2f a  = *(const v2f*)&S0[m][k];    // ds_load_b64, conflict-free (pad 66)
            v2f bm = *(const v2f*)&S1[m][k];
            C = __builtin_amdgcn_wmma_f32_16x16x4_f32(
                    false, a, false, bm, (short)0, C, false, false);
        }

        // Spill C tile to LDS (reuse S0 region; all S0 reads already done,
        // same-wave DS ops are in-order) and pick the diagonal.
        float* cs = &smem[pair][0][0][0];       // 256-float scratch
#pragma unroll
        for (int v = 0; v < 8; ++v) cs[lane * 8 + v] = C[v];

        if (lane < kBatchesPerPair) {
            int n = lane;
            // C element (M,N): lane = N + 16*(M>=8), vgpr = M%8
            // diag n: addr = 9n (n<8) else 9n+120
            int a = (n < 8) ? (9 * n) : (9 * n + 120);
            float sum = cs[a];
            int b = bb + n;
            if (b < B) {
                float x = sum / nb[b];
                out[b] = 1.0f / (1.0f + __expf(-x));
            }
        }
    }
}

extern "C" void kernel_launch(void* const* d_in, const int* in_sizes, int n_in,
                              void* d_out, int out_size, void* d_ws, size_t ws_size,
                              hipStream_t stream) {
    const int*   ii0  = (const int*)d_in[0];
    const int*   ii1  = (const int*)d_in[1];
    const float* vecs = (const float*)d_in[2];
    const float* nb   = (const float*)d_in[3];
    float*       out  = (float*)d_out;
    const int B = in_sizes[3];          // nb_inters has B elements

    const int blocks = (B + kBatchesPerBlock - 1) / kBatchesPerBlock;
    iafm_kernel<<<blocks, kThreads, 0, stream>>>(ii0, ii1, vecs, nb, out, B);
}